// GAT_5995774346005
// MI455X (gfx1250) — compile-verified
//
#include <hip/hip_runtime.h>
#include <math.h>

typedef float v2f __attribute__((ext_vector_type(2)));
typedef float v8f __attribute__((ext_vector_type(8)));

#define FIN 256
#define H1 8
#define D1 8
#define HD1 64
#define KEYNEG 0x007FFFFFu   // order-preserving key of -inf

// ---------- helpers ----------
__device__ __forceinline__ unsigned f2key(float f) {
    unsigned u = __float_as_uint(f);
    return (u & 0x80000000u) ? ~u : (u | 0x80000000u);
}
__device__ __forceinline__ float key2f(unsigned k) {
    unsigned u = (k & 0x80000000u) ? (k & 0x7FFFFFFFu) : ~k;
    return __uint_as_float(u);
}
__device__ __forceinline__ float leaky(float v) { return v > 0.f ? v : 0.2f * v; }

__global__ void fill_u32(unsigned* __restrict__ p, unsigned v, long long n) {
    long long i = (long long)blockIdx.x * blockDim.x + threadIdx.x;
    if (i < n) p[i] = v;
}

// ---------- GEMM1: h1[N,64] = x[N,256] @ W1[256,64]  (f32 WMMA) ----------
__global__ __launch_bounds__(256) void gemm1_wmma(const float* __restrict__ x,
                                                  const float* __restrict__ W1,
                                                  float* __restrict__ h1, int N) {
    int wave = threadIdx.x >> 5;
    int lane = threadIdx.x & 31;
    int lm = lane & 15, lh = lane >> 4;
    int row0 = blockIdx.x * 32 + (wave >> 2) * 16;   // N % 32 == 0
    int col0 = (wave & 3) * 16;
    if (row0 >= N) return;                           // wave-uniform; EXEC stays all-ones
    const float* xr = x + (long long)(row0 + lm) * FIN;
    v8f acc = {0.f, 0.f, 0.f, 0.f, 0.f, 0.f, 0.f, 0.f};
    for (int k0 = 0; k0 < FIN; k0 += 4) {
        int ka = k0 + 2 * lh;
        v2f a, b;
        a.x = xr[ka];
        a.y = xr[ka + 1];
        b.x = W1[ka * HD1 + col0 + lm];
        b.y = W1[(ka + 1) * HD1 + col0 + lm];
        acc = __builtin_amdgcn_wmma_f32_16x16x4_f32(false, a, false, b, (short)0, acc,
                                                    false, false);
    }
    float* out = h1 + (long long)row0 * HD1 + col0 + lm;
#pragma unroll
    for (int k = 0; k < 8; ++k)
        out[(long long)(k + 8 * lh) * HD1] = acc[k];
}

// ---------- alpha1: per (node, head) dot products ----------
__global__ void alpha1_kernel(const float* __restrict__ h1,
                              const float* __restrict__ a_src,
                              const float* __restrict__ a_dst,
                              float* __restrict__ as, float* __restrict__ ad, int N) {
    long long i = (long long)blockIdx.x * blockDim.x + threadIdx.x;  // n*8 + h
    if (i >= (long long)N * H1) return;
    int h = (int)(i & 7);
    const float* hp = h1 + (i >> 3) * HD1 + h * D1;
    float s = 0.f, d = 0.f;
#pragma unroll
    for (int c = 0; c < D1; ++c) {
        float v = hp[c];
        s += v * a_src[h * D1 + c];
        d += v * a_dst[h * D1 + c];
    }
    as[i] = s;
    ad[i] = d;
}

// ---------- layer-1 edge segment-max ----------
__global__ void edge_max1(const int* __restrict__ ei, int E, int N,
                          const float* __restrict__ as, const float* __restrict__ ad,
                          unsigned* __restrict__ emax) {
    long long t = (long long)blockIdx.x * blockDim.x + threadIdx.x;
    long long ET = (long long)E + N;
    if (t >= ET * H1) return;
    int h = (int)(t & 7);
    long long e = t >> 3;
    int s, d;
    if (e < E) { s = ei[e]; d = ei[E + e]; } else { s = d = (int)(e - E); }
    float v = leaky(as[(long long)s * H1 + h] + ad[(long long)d * H1 + h]);
    atomicMax(&emax[(long long)d * H1 + h], f2key(v));
}

// ---------- layer-1 fused edge accumulate: num += w*h1[src], denom += w ----------
__global__ void edge_acc1(const int* __restrict__ ei, int E, int N,
                          const float* __restrict__ as, const float* __restrict__ ad,
                          const unsigned* __restrict__ emax,
                          const float* __restrict__ h1,
                          float* __restrict__ denom, float* __restrict__ num) {
    long long t = (long long)blockIdx.x * blockDim.x + threadIdx.x;
    long long ET = (long long)E + N;
    long long e = t >> 5;                       // one wave (32 lanes) per edge
    if (e >= ET) return;
    int lane = (int)(t & 31);
    int s, d;
    if (e < E) { s = ei[e]; d = ei[E + e]; } else { s = d = (int)(e - E); }
    int h = lane >> 2;                          // 4 lanes per head
    float v = leaky(as[(long long)s * H1 + h] + ad[(long long)d * H1 + h]);
    float w = __expf(v - key2f(emax[(long long)d * H1 + h]));
    if ((lane & 3) == 0) atomicAdd(&denom[(long long)d * H1 + h], w);
    int c0 = lane * 2;                          // two channels of 64 per lane
    const float* hs = h1 + (long long)s * HD1 + c0;
    float* nd = num + (long long)d * HD1 + c0;
    atomicAdd(nd, w * hs[0]);
    atomicAdd(nd + 1, w * hs[1]);
}

// ---------- finalize layer-1: normalize, +bias, ELU -> hx ----------
__global__ void finalize1(const float* __restrict__ num, const float* __restrict__ denom,
                          const float* __restrict__ b1, float* __restrict__ hx, int N) {
    long long i = (long long)blockIdx.x * blockDim.x + threadIdx.x;
    if (i >= (long long)N * HD1) return;
    int hc = (int)(i & 63);
    float v = num[i] / denom[(i >> 6) * H1 + (hc >> 3)] + b1[hc];
    hx[i] = v > 0.f ? v : (__expf(v) - 1.f);
}

// ---------- GEMM2: h2[N,C2] = hx[N,64] @ W2[64,C2]  (f32 WMMA) ----------
__global__ __launch_bounds__(224) void gemm2_wmma(const float* __restrict__ hx,
                                                  const float* __restrict__ W2,
                                                  float* __restrict__ h2, int N, int C2) {
    int wave = threadIdx.x >> 5;   // 7 col tiles (0..6) cover 112 >= C2
    int lane = threadIdx.x & 31;
    int lm = lane & 15, lh = lane >> 4;
    int row0 = blockIdx.x * 16;    // N % 16 == 0
    int col = wave * 16 + lm;
    bool cok = col < C2;
    const float* ar = hx + (long long)(row0 + lm) * HD1;
    v8f acc = {0.f, 0.f, 0.f, 0.f, 0.f, 0.f, 0.f, 0.f};
    for (int k0 = 0; k0 < HD1; k0 += 4) {
        int ka = k0 + 2 * lh;
        v2f a, b;
        a.x = ar[ka];
        a.y = ar[ka + 1];
        b.x = cok ? W2[ka * C2 + col] : 0.f;
        b.y = cok ? W2[(ka + 1) * C2 + col] : 0.f;
        acc = __builtin_amdgcn_wmma_f32_16x16x4_f32(false, a, false, b, (short)0, acc,
                                                    false, false);
    }
    if (cok) {
        float* out = h2 + (long long)row0 * C2 + col;
#pragma unroll
        for (int k = 0; k < 8; ++k)
            out[(long long)(k + 8 * lh) * C2] = acc[k];
    }
}

// ---------- alpha2: wave-per-node reduction over C2 channels ----------
__global__ void alpha2_kernel(const float* __restrict__ h2,
                              const float* __restrict__ a_src,
                              const float* __restrict__ a_dst,
                              float* __restrict__ as, float* __restrict__ ad,
                              int N, int C2) {
    int wid = threadIdx.x >> 5, lane = threadIdx.x & 31;
    int n = blockIdx.x * (blockDim.x >> 5) + wid;
    if (n >= N) return;
    const float* hp = h2 + (long long)n * C2;
    float s = 0.f, d = 0.f;
    for (int c = lane; c < C2; c += 32) {
        float v = hp[c];
        s += v * a_src[c];
        d += v * a_dst[c];
    }
    for (int m = 16; m; m >>= 1) {
        s += __shfl_xor(s, m, 32);
        d += __shfl_xor(d, m, 32);
    }
    if (lane == 0) { as[n] = s; ad[n] = d; }
}

// ---------- layer-2 edge segment-max ----------
__global__ void edge_max2(const int* __restrict__ ei, int E, int N,
                          const float* __restrict__ as, const float* __restrict__ ad,
                          unsigned* __restrict__ emax) {
    long long e = (long long)blockIdx.x * blockDim.x + threadIdx.x;
    long long ET = (long long)E + N;
    if (e >= ET) return;
    int s, d;
    if (e < E) { s = ei[e]; d = ei[E + e]; } else { s = d = (int)(e - E); }
    atomicMax(&emax[d], f2key(leaky(as[s] + ad[d])));
}

// ---------- layer-2 fused edge accumulate (wave per edge) ----------
__global__ void edge_acc2(const int* __restrict__ ei, int E, int N,
                          const float* __restrict__ as, const float* __restrict__ ad,
                          const unsigned* __restrict__ emax,
                          const float* __restrict__ h2,
                          float* __restrict__ denom, float* __restrict__ num, int C2) {
    long long t = (long long)blockIdx.x * blockDim.x + threadIdx.x;
    long long ET = (long long)E + N;
    long long e = t >> 5;
    if (e >= ET) return;
    int lane = (int)(t & 31);
    int s, d;
    if (e < E) { s = ei[e]; d = ei[E + e]; } else { s = d = (int)(e - E); }
    float w = __expf(leaky(as[s] + ad[d]) - key2f(emax[d]));
    if (lane == 0) atomicAdd(&denom[d], w);
    const float* hs = h2 + (long long)s * C2;
    float* nd = num + (long long)d * C2;
    for (int c = lane; c < C2; c += 32)
        atomicAdd(&nd[c], w * hs[c]);
}

// ---------- finalize layer-2: normalize, +bias, sigmoid (in place on d_out) ----------
__global__ void finalize2(float* __restrict__ out, const float* __restrict__ denom,
                          const float* __restrict__ b2, int N, int C2) {
    long long i = (long long)blockIdx.x * blockDim.x + threadIdx.x;
    if (i >= (long long)N * C2) return;
    int c = (int)(i % C2);
    long long n = i / C2;
    float v = out[i] / denom[n] + b2[c];
    out[i] = 1.f / (1.f + __expf(-v));
}

// ---------- host launch ----------
static inline int cdiv(long long a, long long b) { return (int)((a + b - 1) / b); }

extern "C" void kernel_launch(void* const* d_in, const int* in_sizes, int n_in,
                              void* d_out, int out_size, void* d_ws, size_t ws_size,
                              hipStream_t stream) {
    const float* x      = (const float*)d_in[0];
    const int*   ei     = (const int*)d_in[1];
    const float* W1     = (const float*)d_in[2];
    const float* a_src1 = (const float*)d_in[3];
    const float* a_dst1 = (const float*)d_in[4];
    const float* b1     = (const float*)d_in[5];
    const float* W2     = (const float*)d_in[6];
    const float* a_src2 = (const float*)d_in[7];
    const float* a_dst2 = (const float*)d_in[8];
    const float* b2     = (const float*)d_in[9];

    const int N  = in_sizes[0] / FIN;  // 100000
    const int E  = in_sizes[1] / 2;    // 1600000
    const int C2 = in_sizes[9];        // 100
    const long long ET = (long long)E + N;

    // workspace layout (f32 slots)
    float*    h1     = (float*)d_ws;                         // N*64 (reused as hx)
    float*    as1    = h1 + (long long)N * HD1;              // N*8
    float*    ad1    = as1 + (long long)N * H1;              // N*8
    unsigned* emax1  = (unsigned*)(ad1 + (long long)N * H1); // N*8
    float*    denom1 = (float*)(emax1 + (long long)N * H1);  // N*8
    float*    num1   = denom1 + (long long)N * H1;           // N*64
    float*    h2     = num1 + (long long)N * HD1;            // N*C2
    float*    as2    = h2 + (long long)N * C2;               // N
    float*    ad2    = as2 + N;                              // N
    unsigned* emax2  = (unsigned*)(ad2 + N);                 // N
    float*    denom2 = (float*)(emax2 + N);                  // N
    float*    out    = (float*)d_out;                        // num2 accum + result

    const int B = 256;

    // ---- init accumulators ----
    fill_u32<<<cdiv((long long)N * H1, B), B, 0, stream>>>(emax1, KEYNEG, (long long)N * H1);
    fill_u32<<<cdiv((long long)N * (H1 + HD1), B), B, 0, stream>>>(
        (unsigned*)denom1, 0u, (long long)N * (H1 + HD1));   // denom1 + num1 contiguous
    fill_u32<<<cdiv(N, B), B, 0, stream>>>(emax2, KEYNEG, N);
    fill_u32<<<cdiv(N, B), B, 0, stream>>>((unsigned*)denom2, 0u, N);
    fill_u32<<<cdiv((long long)N * C2, B), B, 0, stream>>>((unsigned*)out, 0u,
                                                           (long long)N * C2);

    // ---- layer 1 ----
    gemm1_wmma<<<cdiv(N, 32), 256, 0, stream>>>(x, W1, h1, N);
    alpha1_kernel<<<cdiv((long long)N * H1, B), B, 0, stream>>>(h1, a_src1, a_dst1,
                                                                as1, ad1, N);
    edge_max1<<<cdiv(ET * H1, B), B, 0, stream>>>(ei, E, N, as1, ad1, emax1);
    edge_acc1<<<cdiv(ET * 32, B), B, 0, stream>>>(ei, E, N, as1, ad1, emax1, h1,
                                                  denom1, num1);
    finalize1<<<cdiv((long long)N * HD1, B), B, 0, stream>>>(num1, denom1, b1, h1, N);

    // ---- layer 2 ----
    gemm2_wmma<<<cdiv(N, 16), 224, 0, stream>>>(h1, W2, h2, N, C2);
    alpha2_kernel<<<cdiv(N, 8), 256, 0, stream>>>(h2, a_src2, a_dst2, as2, ad2, N, C2);
    edge_max2<<<cdiv(ET, B), B, 0, stream>>>(ei, E, N, as2, ad2, emax2);
    edge_acc2<<<cdiv(ET * 32, B), B, 0, stream>>>(ei, E, N, as2, ad2, emax2, h2,
                                                  denom2, out, C2);
    finalize2<<<cdiv((long long)N * C2, B), B, 0, stream>>>(out, denom2, b2, N, C2);
}